// AttnDecoderRNN_5171140624839
// MI455X (gfx1250) — compile-verified
//
#include <hip/hip_runtime.h>
#include <hip/hip_bf16.h>

// Sizes from the reference
#define VSZ 50257
#define ESZ 1024
#define HSZ 1024
#define LSZ 52

typedef __attribute__((ext_vector_type(2))) float v2f;
typedef __attribute__((ext_vector_type(8))) float v8f;

// Workspace layout (float offsets)
#define OFF_XCAT 0            // [2048]  concat(emb, attn_applied)
#define OFF_XG   2048         // [1024]  relu(comb(...))
#define OFF_GI   3072         // [3072]
#define OFF_GH   6144         // [3072]
#define OFF_LOG  9216         // [50257] logits
#define OFF_HNEW 59474        // [1024]  h_new (8B aligned copy)
#define OFF_RED  60498        // [2]     {max, log(sum exp)}

// ---------------------------------------------------------------------------
// One wave computes y[row0 .. row0+15] = W[rows, 0:K] . x (+bias, opt. relu)
// via V_WMMA_F32_16X16X4_F32 accumulation. K must be a multiple of 4 and the
// bases 8-byte aligned (true for all uses here).
//   A 16x4 layout : lanes 0-15 -> (M=lane, K=k..k+1), lanes 16-31 -> K=k+2..k+3
//   B  4x16 layout: only column N=0 nonzero (lane 0 / lane 16 carry x chunk)
//   D column 0    : lane 0 holds M=0..7 in acc[0..7], lane 16 holds M=8..15
// ---------------------------------------------------------------------------
__device__ __forceinline__ void wmma_gemv_wave(
    const float* __restrict__ W, const float* __restrict__ x,
    const float* __restrict__ bias, float* __restrict__ y,
    int row0, int nrows, int K, bool do_relu)
{
  const int lane = threadIdx.x & 31;
  const int m    = lane & 15;            // row within 16-row group
  const int kh   = (lane >> 4) << 1;     // 0 (lanes 0-15) or 2 (lanes 16-31)

  int row = row0 + m;
  if (row >= nrows) row = nrows - 1;     // clamp; extra rows are never stored
  const float* wrow = W + (size_t)row * (size_t)K;

  v8f acc = {};
  #pragma unroll 4
  for (int k = 0; k < K; k += 4) {
    v2f a  = *(const v2f*)(wrow + k + kh);   // 16 rows x 4 K of W, read once
    v2f xv = *(const v2f*)(x + k + kh);      // broadcast chunk of x
    v2f b;
    b.x = (m == 0) ? xv.x : 0.0f;            // only N=0 column of B is live
    b.y = (m == 0) ? xv.y : 0.0f;
    acc = __builtin_amdgcn_wmma_f32_16x16x4_f32(
        /*neg_a=*/false, a, /*neg_b=*/false, b,
        /*c_mod=*/(short)0, acc, /*reuse_a=*/false, /*reuse_b=*/false);
  }

  if (m == 0) {                              // lanes 0 and 16 hold column N=0
    const int base = row0 + ((lane >> 4) << 3);
    #pragma unroll
    for (int i = 0; i < 8; ++i) {
      int r = base + i;
      if (r < nrows) {
        float v = acc[i] + (bias ? bias[r] : 0.0f);
        if (do_relu) v = v > 0.0f ? v : 0.0f;
        y[r] = v;
      }
    }
  }
}

// Generic GEMV kernel: 256 threads = 8 waves = 128 rows per block.
__global__ void k_wmma_gemv(const float* __restrict__ W, const float* __restrict__ x,
                            const float* __restrict__ bias, float* __restrict__ y,
                            int nrows, int K, int do_relu)
{
  int wave = (int)((blockIdx.x * blockDim.x + threadIdx.x) >> 5);
  int row0 = wave * 16;
  if (row0 >= nrows) return;               // wave-uniform exit, EXEC stays full
  wmma_gemv_wave(W, x, bias, y, row0, nrows, K, do_relu != 0);
}

// Embedding gather + linear GEMV (x = emb_table[token])
__global__ void k_embed_linear(const int* __restrict__ token,
                               const float* __restrict__ emb_table,
                               const float* __restrict__ lin_w,
                               const float* __restrict__ lin_b,
                               float* __restrict__ y)
{
  const float* x = emb_table + (size_t)token[0] * (size_t)ESZ;
  int wave = (int)((blockIdx.x * blockDim.x + threadIdx.x) >> 5);
  int row0 = wave * 16;
  if (row0 >= HSZ) return;
  wmma_gemv_wave(lin_w, x, lin_b, y, row0, HSZ, ESZ, false);
}

// Attention: logits over L=52, softmax, weighted sum of encoder outputs.
// Single block of 256 threads (8 waves).
__global__ void k_attn(const float* __restrict__ xcat,      // emb in [0,1024)
                       const float* __restrict__ hidden,    // h0 [1024]
                       const float* __restrict__ enc,       // [52,1024]
                       const float* __restrict__ attn_w,    // [52,2048]
                       const float* __restrict__ attn_b,    // [52]
                       float* __restrict__ attn_applied,    // ws + 1024
                       float* __restrict__ attn_weights_out)// d_out + V + H
{
  __shared__ float s_log[64];
  __shared__ float s_w[64];
  const int tid = threadIdx.x, wave = tid >> 5, lane = tid & 31;

  for (int l = wave; l < LSZ; l += 8) {
    const float* wr = attn_w + l * (2 * HSZ);
    float s = 0.0f;
    for (int k = lane; k < HSZ; k += 32) s += wr[k] * xcat[k];
    for (int k = lane; k < HSZ; k += 32) s += wr[HSZ + k] * hidden[k];
    for (int off = 16; off; off >>= 1) s += __shfl_xor(s, off, 32);
    if (lane == 0) s_log[l] = s + attn_b[l];
  }
  __syncthreads();

  if (tid == 0) {
    float mx = -3.0e38f;
    for (int l = 0; l < LSZ; ++l) mx = fmaxf(mx, s_log[l]);
    float sum = 0.0f;
    for (int l = 0; l < LSZ; ++l) { float e = __expf(s_log[l] - mx); s_w[l] = e; sum += e; }
    float inv = 1.0f / sum;
    for (int l = 0; l < LSZ; ++l) s_w[l] *= inv;
  }
  __syncthreads();

  if (tid < LSZ) attn_weights_out[tid] = s_w[tid];
  for (int c = tid; c < HSZ; c += 256) {
    float s = 0.0f;
    for (int l = 0; l < LSZ; ++l) s += s_w[l] * enc[l * HSZ + c];
    attn_applied[c] = s;
  }
}

// GRU gate math (PyTorch GRUCell equations)
__global__ void k_gates(const float* __restrict__ gi, const float* __restrict__ gh,
                        const float* __restrict__ hidden,
                        float* __restrict__ hnew_out,  // d_out + V
                        float* __restrict__ hnew_ws)   // aligned ws copy
{
  int i = (int)(blockIdx.x * blockDim.x + threadIdx.x);
  if (i >= HSZ) return;
  float r = 1.0f / (1.0f + __expf(-(gi[i] + gh[i])));
  float z = 1.0f / (1.0f + __expf(-(gi[HSZ + i] + gh[HSZ + i])));
  float n = tanhf(gi[2 * HSZ + i] + r * gh[2 * HSZ + i]);
  float h = (1.0f - z) * n + z * hidden[i];
  hnew_out[i] = h;
  hnew_ws[i]  = h;
}

// Single-block reduction over the vocab: max and log(sum exp).
__global__ void k_lse(const float* __restrict__ logits, float* __restrict__ red)
{
  __shared__ float s[1024];
  const int tid = threadIdx.x;
  float mx = -3.0e38f;
  for (int v = tid; v < VSZ; v += 1024) mx = fmaxf(mx, logits[v]);
  s[tid] = mx; __syncthreads();
  for (int off = 512; off; off >>= 1) {
    if (tid < off) s[tid] = fmaxf(s[tid], s[tid + off]);
    __syncthreads();
  }
  float m = s[0];
  __syncthreads();
  float sum = 0.0f;
  for (int v = tid; v < VSZ; v += 1024) sum += __expf(logits[v] - m);
  s[tid] = sum; __syncthreads();
  for (int off = 512; off; off >>= 1) {
    if (tid < off) s[tid] += s[tid + off];
    __syncthreads();
  }
  if (tid == 0) { red[0] = m; red[1] = __logf(s[0]); }
}

__global__ void k_logsm(const float* __restrict__ logits, const float* __restrict__ red,
                        float* __restrict__ out)
{
  int v = (int)(blockIdx.x * blockDim.x + threadIdx.x);
  if (v < VSZ) out[v] = logits[v] - red[0] - red[1];
}

extern "C" void kernel_launch(void* const* d_in, const int* in_sizes, int n_in,
                              void* d_out, int out_size, void* d_ws, size_t ws_size,
                              hipStream_t stream)
{
  const int*   token  = (const int*)  d_in[0];
  const float* hidden = (const float*)d_in[1];
  const float* enc    = (const float*)d_in[2];
  const float* embT   = (const float*)d_in[3];
  const float* lin_w  = (const float*)d_in[4];
  const float* lin_b  = (const float*)d_in[5];
  const float* attn_w = (const float*)d_in[6];
  const float* attn_b = (const float*)d_in[7];
  const float* comb_w = (const float*)d_in[8];
  const float* comb_b = (const float*)d_in[9];
  const float* gwih   = (const float*)d_in[10];
  const float* gwhh   = (const float*)d_in[11];
  const float* gbih   = (const float*)d_in[12];
  const float* gbhh   = (const float*)d_in[13];
  const float* out_w  = (const float*)d_in[14];
  const float* out_b  = (const float*)d_in[15];

  float* out  = (float*)d_out;             // [V] log_probs | [H] h_new | [L] attn_w
  float* ws   = (float*)d_ws;
  float* xcat = ws + OFF_XCAT;
  float* xg   = ws + OFF_XG;
  float* gi   = ws + OFF_GI;
  float* gh   = ws + OFF_GH;
  float* lgts = ws + OFF_LOG;
  float* hnw  = ws + OFF_HNEW;
  float* red  = ws + OFF_RED;

  const int TB = 256;               // 8 waves -> 128 rows per block
  const int ROWS_PER_BLOCK = 128;

  // 1) emb = linear(emb_table[token])  -> xcat[0:1024)
  k_embed_linear<<<HSZ / ROWS_PER_BLOCK, TB, 0, stream>>>(token, embT, lin_w, lin_b, xcat);

  // 2) attention -> xcat[1024:2048) and attn_weights -> d_out tail
  k_attn<<<1, TB, 0, stream>>>(xcat, hidden, enc, attn_w, attn_b,
                               xcat + HSZ, out + VSZ + HSZ);

  // 3) x = relu(comb(concat))  (1024 x 2048)
  k_wmma_gemv<<<HSZ / ROWS_PER_BLOCK, TB, 0, stream>>>(comb_w, xcat, comb_b, xg,
                                                       HSZ, 2 * HSZ, 1);

  // 4) GRU input/hidden GEMVs (3072 x 1024 each)
  k_wmma_gemv<<<(3 * HSZ) / ROWS_PER_BLOCK, TB, 0, stream>>>(gwih, xg, gbih, gi,
                                                             3 * HSZ, HSZ, 0);
  k_wmma_gemv<<<(3 * HSZ) / ROWS_PER_BLOCK, TB, 0, stream>>>(gwhh, hidden, gbhh, gh,
                                                             3 * HSZ, HSZ, 0);

  // 5) gates -> h_new (d_out + V, plus aligned ws copy)
  k_gates<<<(HSZ + TB - 1) / TB, TB, 0, stream>>>(gi, gh, hidden, out + VSZ, hnw);

  // 6) logits = out_w @ h_new + out_b  (50257 x 1024) -- the 206MB streamer
  k_wmma_gemv<<<(VSZ + ROWS_PER_BLOCK - 1) / ROWS_PER_BLOCK, TB, 0, stream>>>(
      out_w, hnw, out_b, lgts, VSZ, HSZ, 0);

  // 7) log_softmax
  k_lse<<<1, 1024, 0, stream>>>(lgts, red);
  k_logsm<<<(VSZ + TB - 1) / TB, TB, 0, stream>>>(lgts, red, out);
}